// S5Layer_5987184411214
// MI455X (gfx1250) — compile-verified
//
#include <hip/hip_runtime.h>

// S5 layer: L=16384, H=512, P=512, fp32 end-to-end using CDNA5 fp32 WMMA,
// async global->LDS staging for the shared weight panels, chunked scan.

#define L_SEQ   16384
#define H_DIM   512
#define P_DIM   512
#define CHUNK   128
#define NCHUNKS (L_SEQ / CHUNK)   // 128
#define BPAD    516               // LDS row stride (floats): 516 mod 64 = 4 -> conflict-free b64 reads

typedef float v2f __attribute__((ext_vector_type(2)));
typedef float v8f __attribute__((ext_vector_type(8)));

// ---- CDNA5 async global->LDS copy (ASYNCcnt-tracked, per ISA 10. / 15.18.3) ----
__device__ __forceinline__ void async_copy_b128(uint32_t lds_off, const float* gptr) {
  asm volatile("global_load_async_to_lds_b128 %0, %1, off"
               :: "v"(lds_off), "v"(gptr)
               : "memory");
}
__device__ __forceinline__ void wait_async_all() {
  asm volatile("s_wait_asynccnt 0x0" ::: "memory");
}

// ---------------------------------------------------------------------------
// K0: discretization + weight de-interleave.
// ---------------------------------------------------------------------------
__global__ __launch_bounds__(256) void s5_k0_prep(
    const float* __restrict__ Lre, const float* __restrict__ Lim,
    const float* __restrict__ Bin, const float* __restrict__ Cin,
    const float* __restrict__ logstep,
    float* __restrict__ Bbar_re, float* __restrict__ Bbar_im,
    float* __restrict__ Cre, float* __restrict__ Cim,
    float* __restrict__ lb_re, float* __restrict__ lb_im) {
  int idx = blockIdx.x * blockDim.x + threadIdx.x;
  if (idx >= P_DIM * H_DIM) return;
  int p = idx / H_DIM;
  int h = idx % H_DIM;

  float dt = expf(logstep[p]);
  float lr = Lre[p], li = Lim[p];
  float er  = expf(lr * dt);
  float ang = li * dt;
  float lbr = er * cosf(ang);
  float lbi = er * sinf(ang);
  float nr  = lbr - 1.0f, ni = lbi;
  float inv = 1.0f / (lr * lr + li * li);
  float cr  = (nr * lr + ni * li) * inv;
  float ci  = (ni * lr - nr * li) * inv;
  float br = Bin[(size_t)idx * 2 + 0];
  float bi = Bin[(size_t)idx * 2 + 1];
  Bbar_re[idx] = cr * br - ci * bi;
  Bbar_im[idx] = cr * bi + ci * br;
  Cre[idx] = Cin[(size_t)idx * 2 + 0];   // idx == h*P+p over the same range
  Cim[idx] = Cin[(size_t)idx * 2 + 1];
  if (h == 0) { lb_re[p] = lbr; lb_im[p] = lbi; }
}

// ---------------------------------------------------------------------------
// K1: Bu = U @ B_bar^T via fp32 WMMA. One wave = 4 M-tiles x 1 N-tile,
// re+im accumulators (8x v8f). 16xH B-panel (re,im) async-staged in LDS and
// shared by all 8 waves of the block. 6 loads / 8 WMMAs per K-step.
// ---------------------------------------------------------------------------
__global__ __launch_bounds__(256) void s5_k1_gemm_bu(
    const float* __restrict__ U,
    const float* __restrict__ Bre, const float* __restrict__ Bim,
    float* __restrict__ Bu_re, float* __restrict__ Bu_im) {
  __shared__ float sB[2][16 * BPAD];
  const int tilesN = P_DIM / 16;               // 32
  const int blkM   = blockIdx.x / tilesN;      // 0..31 (each covers 512 rows)
  const int tn     = blockIdx.x % tilesN;
  const int tid    = threadIdx.x;

  // Stage 16x512 re + im panels (64KB) with async b128 copies.
  {
    const float* src0 = Bre + (size_t)tn * 16 * H_DIM;
    const float* src1 = Bim + (size_t)tn * 16 * H_DIM;
#pragma unroll
    for (int it = 0; it < 8; ++it) {
      int chunk = it * 256 + tid;              // 0..2047 float4 chunks
      int row = chunk >> 7;
      int col = (chunk & 127) << 2;
      uint32_t l0 = (uint32_t)(uintptr_t)&sB[0][row * BPAD + col];
      uint32_t l1 = (uint32_t)(uintptr_t)&sB[1][row * BPAD + col];
      async_copy_b128(l0, src0 + (size_t)row * H_DIM + col);
      async_copy_b128(l1, src1 + (size_t)row * H_DIM + col);
    }
    wait_async_all();
    __syncthreads();
  }

  const int lane = tid & 31, wave = tid >> 5;
  const int g = lane >> 4, mr = lane & 15;
  const int rowBase = (blkM * 8 + wave) * 64;  // 4 M-tiles of 16 rows

  const float* aBase = U + (size_t)(rowBase + mr) * H_DIM + 2 * g;
  const float* bR = &sB[0][mr * BPAD + 2 * g];
  const float* bI = &sB[1][mr * BPAD + 2 * g];

  v8f accR[4] = {};
  v8f accI[4] = {};
#pragma unroll 2
  for (int k0 = 0; k0 < H_DIM; k0 += 4) {
    v2f br = *reinterpret_cast<const v2f*>(bR + k0);
    v2f bi = *reinterpret_cast<const v2f*>(bI + k0);
#pragma unroll
    for (int t = 0; t < 4; ++t) {
      v2f a = *reinterpret_cast<const v2f*>(aBase + (size_t)t * 16 * H_DIM + k0);
      accR[t] = __builtin_amdgcn_wmma_f32_16x16x4_f32(false, a, false, br,
                                                      (short)0, accR[t], false, false);
      accI[t] = __builtin_amdgcn_wmma_f32_16x16x4_f32(false, a, false, bi,
                                                      (short)0, accI[t], false, false);
    }
  }
#pragma unroll
  for (int t = 0; t < 4; ++t) {
#pragma unroll
    for (int r = 0; r < 8; ++r) {
      int row = rowBase + t * 16 + g * 8 + r;
      int col = tn * 16 + mr;
      Bu_re[(size_t)row * P_DIM + col] = accR[t][r];
      Bu_im[(size_t)row * P_DIM + col] = accI[t][r];
    }
  }
}

// ---------------------------------------------------------------------------
// K2a: read-only pass -> per-chunk carries (local scan end state).
// ---------------------------------------------------------------------------
__global__ __launch_bounds__(P_DIM) void s5_k2a_chunk_carry(
    const float* __restrict__ Bu_re, const float* __restrict__ Bu_im,
    const float* __restrict__ lb_re, const float* __restrict__ lb_im,
    float* __restrict__ carry_re, float* __restrict__ carry_im) {
  const int p = threadIdx.x;
  const int c = blockIdx.x;
  const float lr = lb_re[p], li = lb_im[p];
  float xr = 0.0f, xi = 0.0f;
  size_t base = (size_t)c * CHUNK * P_DIM + p;
  for (int t = 0; t < CHUNK; ++t) {
    size_t idx = base + (size_t)t * P_DIM;
    float br = Bu_re[idx], bi = Bu_im[idx];
    float nr = lr * xr - li * xi + br;
    float ni = lr * xi + li * xr + bi;
    xr = nr; xi = ni;
  }
  carry_re[c * P_DIM + p] = xr;
  carry_im[c * P_DIM + p] = xi;
}

// ---------------------------------------------------------------------------
// K2b: sequential cross-chunk combine -> exclusive carries.
// ---------------------------------------------------------------------------
__global__ __launch_bounds__(P_DIM) void s5_k2b_carry_scan(
    const float* __restrict__ carry_re, const float* __restrict__ carry_im,
    float* __restrict__ cex_re, float* __restrict__ cex_im,
    const float* __restrict__ lb_re, const float* __restrict__ lb_im) {
  const int p = threadIdx.x;
  const float lr = lb_re[p], li = lb_im[p];
  float tr = 1.0f, ti = 0.0f;                  // lb^CHUNK
  for (int i = 0; i < CHUNK; ++i) {
    float nr = tr * lr - ti * li;
    float ni = tr * li + ti * lr;
    tr = nr; ti = ni;
  }
  float er = 0.0f, ei = 0.0f;
  for (int c = 0; c < NCHUNKS; ++c) {
    cex_re[c * P_DIM + p] = er;
    cex_im[c * P_DIM + p] = ei;
    float cr = carry_re[c * P_DIM + p];
    float ci = carry_im[c * P_DIM + p];
    float nr = tr * er - ti * ei + cr;
    float ni = tr * ei + ti * er + ci;
    er = nr; ei = ni;
  }
}

// ---------------------------------------------------------------------------
// K2c: full scan with carry-in, writes xs in place (single RW pass).
// ---------------------------------------------------------------------------
__global__ __launch_bounds__(P_DIM) void s5_k2c_scan(
    float* __restrict__ Bu_re, float* __restrict__ Bu_im,
    const float* __restrict__ cex_re, const float* __restrict__ cex_im,
    const float* __restrict__ lb_re, const float* __restrict__ lb_im) {
  const int p = threadIdx.x;
  const int c = blockIdx.x;
  const float lr = lb_re[p], li = lb_im[p];
  float xr = cex_re[c * P_DIM + p];
  float xi = cex_im[c * P_DIM + p];
  size_t base = (size_t)c * CHUNK * P_DIM + p;
  for (int t = 0; t < CHUNK; ++t) {
    size_t idx = base + (size_t)t * P_DIM;
    float br = Bu_re[idx], bi = Bu_im[idx];
    float nr = lr * xr - li * xi + br;
    float ni = lr * xi + li * xr + bi;
    xr = nr; xi = ni;
    Bu_re[idx] = xr;
    Bu_im[idx] = xi;
  }
}

// ---------------------------------------------------------------------------
// K3: ys = 2*(Xre @ Cre^T - Xim @ Cim^T) + D*U. One wave = 4 M-tiles;
// C panels async-staged in LDS. Imag term negated on the A fragment
// (f32 WMMA has no A/B NEG modifier per ISA).
// ---------------------------------------------------------------------------
__global__ __launch_bounds__(256) void s5_k3_gemm_out(
    const float* __restrict__ Xre, const float* __restrict__ Xim,
    const float* __restrict__ Cre, const float* __restrict__ Cim,
    const float* __restrict__ U, const float* __restrict__ D,
    float* __restrict__ out) {
  __shared__ float sC[2][16 * BPAD];
  const int tilesN = H_DIM / 16;               // 32
  const int blkM   = blockIdx.x / tilesN;
  const int tn     = blockIdx.x % tilesN;
  const int tid    = threadIdx.x;

  {
    const float* src0 = Cre + (size_t)tn * 16 * P_DIM;
    const float* src1 = Cim + (size_t)tn * 16 * P_DIM;
#pragma unroll
    for (int it = 0; it < 8; ++it) {
      int chunk = it * 256 + tid;
      int row = chunk >> 7;
      int col = (chunk & 127) << 2;
      uint32_t l0 = (uint32_t)(uintptr_t)&sC[0][row * BPAD + col];
      uint32_t l1 = (uint32_t)(uintptr_t)&sC[1][row * BPAD + col];
      async_copy_b128(l0, src0 + (size_t)row * P_DIM + col);
      async_copy_b128(l1, src1 + (size_t)row * P_DIM + col);
    }
    wait_async_all();
    __syncthreads();
  }

  const int lane = tid & 31, wave = tid >> 5;
  const int g = lane >> 4, mr = lane & 15;
  const int rowBase = (blkM * 8 + wave) * 64;

  const float* arBase = Xre + (size_t)(rowBase + mr) * P_DIM + 2 * g;
  const float* aiBase = Xim + (size_t)(rowBase + mr) * P_DIM + 2 * g;
  const float* bR = &sC[0][mr * BPAD + 2 * g];
  const float* bI = &sC[1][mr * BPAD + 2 * g];

  v8f acc[4] = {};
#pragma unroll 2
  for (int k0 = 0; k0 < P_DIM; k0 += 4) {
    v2f br = *reinterpret_cast<const v2f*>(bR + k0);
    v2f bi = *reinterpret_cast<const v2f*>(bI + k0);
#pragma unroll
    for (int t = 0; t < 4; ++t) {
      v2f ar = *reinterpret_cast<const v2f*>(arBase + (size_t)t * 16 * P_DIM + k0);
      acc[t] = __builtin_amdgcn_wmma_f32_16x16x4_f32(false, ar, false, br,
                                                     (short)0, acc[t], false, false);
      v2f ai = *reinterpret_cast<const v2f*>(aiBase + (size_t)t * 16 * P_DIM + k0);
      ai = -ai;
      acc[t] = __builtin_amdgcn_wmma_f32_16x16x4_f32(false, ai, false, bi,
                                                     (short)0, acc[t], false, false);
    }
  }
#pragma unroll
  for (int t = 0; t < 4; ++t) {
#pragma unroll
    for (int r = 0; r < 8; ++r) {
      int row = rowBase + t * 16 + g * 8 + r;
      int col = tn * 16 + mr;
      size_t o = (size_t)row * H_DIM + col;
      out[o] = 2.0f * acc[t][r] + D[col] * U[o];
    }
  }
}

// ---------------------------------------------------------------------------
extern "C" void kernel_launch(void* const* d_in, const int* in_sizes, int n_in,
                              void* d_out, int out_size, void* d_ws, size_t ws_size,
                              hipStream_t stream) {
  const float* U       = (const float*)d_in[0];  // (L,H)
  const float* Lre     = (const float*)d_in[1];  // (P,)
  const float* Lim     = (const float*)d_in[2];  // (P,)
  const float* Bin     = (const float*)d_in[3];  // (P,H,2)
  const float* Cin     = (const float*)d_in[4];  // (H,P,2)
  const float* D       = (const float*)d_in[5];  // (H,)
  const float* logstep = (const float*)d_in[6];  // (P,)
  float* out = (float*)d_out;                    // (L,H)

  float* ws = (float*)d_ws;
  const size_t PH = (size_t)P_DIM * H_DIM;
  float* Bbar_re  = ws;
  float* Bbar_im  = Bbar_re + PH;
  float* Cre      = Bbar_im + PH;
  float* Cim      = Cre + PH;
  float* lb_re    = Cim + PH;
  float* lb_im    = lb_re + P_DIM;
  float* carry_re = lb_im + P_DIM;
  float* carry_im = carry_re + (size_t)NCHUNKS * P_DIM;
  float* cex_re   = carry_im + (size_t)NCHUNKS * P_DIM;
  float* cex_im   = cex_re + (size_t)NCHUNKS * P_DIM;
  float* Bu_re    = cex_im + (size_t)NCHUNKS * P_DIM;
  float* Bu_im    = Bu_re + (size_t)L_SEQ * P_DIM;

  s5_k0_prep<<<(P_DIM * H_DIM) / 256, 256, 0, stream>>>(
      Lre, Lim, Bin, Cin, logstep, Bbar_re, Bbar_im, Cre, Cim, lb_re, lb_im);

  // 32 M-panels (512 rows each) x 32 N-tiles
  s5_k1_gemm_bu<<<(L_SEQ / 512) * (P_DIM / 16), 256, 0, stream>>>(
      U, Bbar_re, Bbar_im, Bu_re, Bu_im);

  s5_k2a_chunk_carry<<<NCHUNKS, P_DIM, 0, stream>>>(
      Bu_re, Bu_im, lb_re, lb_im, carry_re, carry_im);
  s5_k2b_carry_scan<<<1, P_DIM, 0, stream>>>(
      carry_re, carry_im, cex_re, cex_im, lb_re, lb_im);
  s5_k2c_scan<<<NCHUNKS, P_DIM, 0, stream>>>(
      Bu_re, Bu_im, cex_re, cex_im, lb_re, lb_im);

  s5_k3_gemm_out<<<(L_SEQ / 512) * (H_DIM / 16), 256, 0, stream>>>(
      Bu_re, Bu_im, Cre, Cim, U, D, out);
}